// SelfModifyingTitans_25451976196905
// MI455X (gfx1250) — compile-verified
//
#include <hip/hip_runtime.h>

// SelfModifyingTitans (delta-rule scan) for MI455X / gfx1250.
// Pipeline: [proj: q,k,v = x@W*] -> [chunked WY-exact delta-rule scan, W in LDS,
// all GEMMs via v_wmma_f32_16x16x4_f32, chunk tiles double-buffered via
// GLOBAL_LOAD_ASYNC_TO_LDS_B128] -> [outproj: out = Ys@Wo].
//
// Workspace layout (fp32): Q | K | V | Ys, each B*S*HD = 2,097,152 floats (32 MB total).

#define B_   4
#define S_   4096
#define D_   1024
#define HD_  128
#define CH_  16                 // chunk length (WY chunk)
#define NCH_ (S_ / CH_)         // 256 chunks

typedef __attribute__((ext_vector_type(2))) float v2f;
typedef __attribute__((ext_vector_type(8))) float v8f;

__device__ __forceinline__ v8f wmma4(v2f a, v2f b, v8f c) {
  // D = A(16x4,f32) * B(4x16,f32) + C(16x16,f32)
  return __builtin_amdgcn_wmma_f32_16x16x4_f32(false, a, false, b, (short)0, c,
                                               false, false);
}

// Async DMA: 16 bytes per lane, global -> LDS, tracked by ASYNCcnt.
__device__ __forceinline__ void async_copy_b128(uint32_t lds_addr, const void* gptr) {
  asm volatile("global_load_async_to_lds_b128 %0, %1, off"
               :: "v"(lds_addr), "v"(gptr)
               : "memory");
}

__device__ __forceinline__ void wait_async0() {
#if __has_builtin(__builtin_amdgcn_s_wait_asynccnt)
  __builtin_amdgcn_s_wait_asynccnt(0);
#else
  asm volatile("s_wait_asynccnt 0x0" ::: "memory");
#endif
}

// Low 32 bits of a generic LDS pointer == byte offset in the wave's LDS window.
__device__ __forceinline__ uint32_t lds_addr_of(const void* p) {
  return (uint32_t)(uintptr_t)p;
}

// ---------------------------------------------------------------------------
// Kernel 1: Q/K/V projections.  grid = (M/128, 3), block = 256 (8 waves).
// Wave wv computes a 16-row M-tile across all 8 N-tiles (HD=128) for one matrix.
// ---------------------------------------------------------------------------
__global__ void titans_proj(const float* __restrict__ x,
                            const float* __restrict__ Wq,
                            const float* __restrict__ Wk,
                            const float* __restrict__ Wv,
                            float* __restrict__ Q,
                            float* __restrict__ K,
                            float* __restrict__ V) {
  const int lane = threadIdx.x & 31;
  const int wv   = threadIdx.x >> 5;
  const int half = lane >> 4;
  const int lr   = lane & 15;

  const int mat = blockIdx.y;
  const float* __restrict__ Wm = (mat == 0) ? Wq : (mat == 1) ? Wk : Wv;
  float* __restrict__ Out      = (mat == 0) ? Q  : (mat == 1) ? K  : V;

  const size_t row0 = (size_t)blockIdx.x * 128 + (size_t)wv * 16;
  const size_t rA   = row0 + (size_t)lr;

  v8f acc[8];
#pragma unroll
  for (int i = 0; i < 8; ++i) { v8f z = {}; acc[i] = z; }

  for (int k0 = 0; k0 < D_; k0 += 4) {
    v2f a;
    a.x = x[rA * D_ + (size_t)(k0 + 2 * half + 0)];
    a.y = x[rA * D_ + (size_t)(k0 + 2 * half + 1)];
#pragma unroll
    for (int nt = 0; nt < 8; ++nt) {
      v2f b;
      b.x = Wm[(size_t)(k0 + 2 * half + 0) * HD_ + nt * 16 + lr];
      b.y = Wm[(size_t)(k0 + 2 * half + 1) * HD_ + nt * 16 + lr];
      acc[nt] = wmma4(a, b, acc[nt]);
    }
  }
#pragma unroll
  for (int nt = 0; nt < 8; ++nt)
#pragma unroll
    for (int r = 0; r < 8; ++r)
      Out[(row0 + (size_t)(r + 8 * half)) * HD_ + nt * 16 + lr] = acc[nt][r];
}

// ---------------------------------------------------------------------------
// Kernel 2: chunked delta-rule scan.  grid = B_, block = 256 (8 waves).
// Wave wv owns W rows [16*wv, 16*wv+16).  Per chunk (exact WY form):
//   Yq0 = W Qc^T, Yk0 = W Kc^T        (WMMA, K=128)
//   Ag = Kc Kc^T, Mg = Kc Qc^T        (WMMA, waves 0/1)
//   forward substitution (16 steps, VALU) -> Err, y_t
//   W -= eta * Err Kc                 (WMMA rank-16 update)
// Chunk Q/K/V tiles are double-buffered in LDS; the next chunk's tiles are
// fetched with GLOBAL_LOAD_ASYNC_TO_LDS_B128 concurrently with compute,
// taking the global-load latency off the 256-deep serial chain.
// ---------------------------------------------------------------------------
__global__ void titans_scan(const float* __restrict__ Q,
                            const float* __restrict__ K,
                            const float* __restrict__ V,
                            const float* __restrict__ state,
                            const float* __restrict__ lr,
                            float* __restrict__ Ys,
                            float* __restrict__ Wfinal) {
  extern __shared__ float smem[];
  float* Wsh  = smem;                      // 128*128
  float* cbuf = Wsh + HD_ * HD_;           // 2 x (Qc|Kc|Vc), each 16*128
  float* Ag   = cbuf + 2 * 3 * CH_ * HD_;  // 16*16  Ag[s][t] = k_s . k_t
  float* Mg   = Ag + CH_ * CH_;            // 16*16  Mg[s][t] = k_s . q_t
  float* yqb  = Mg + CH_ * CH_;            // 8 * 16*16 per-wave Yq0 slices
  float* ykb  = yqb + 8 * CH_ * CH_;       // 8 * 16*16
  float* erb  = ykb + 8 * CH_ * CH_;       // 8 * 16*16 (-eta * err)

  const int b    = blockIdx.x;
  const int tid  = threadIdx.x;
  const int lane = tid & 31;
  const int wv   = tid >> 5;
  const int half = lane >> 4;
  const int lr16 = lane & 15;
  const int row0 = wv * 16;
  const float eta = lr[0];

  const int NF4 = CH_ * HD_ / 4;  // 512 float4 per 16x128 tile

  // Issue async prefetch of chunk 0 into buffer 0 before anything else.
  {
    const float4* qg = (const float4*)(Q + (size_t)b * S_ * HD_);
    const float4* kg = (const float4*)(K + (size_t)b * S_ * HD_);
    const float4* vg = (const float4*)(V + (size_t)b * S_ * HD_);
    const uint32_t base = lds_addr_of(cbuf);
    for (int i = tid; i < NF4; i += 256) {
      async_copy_b128(base + (uint32_t)(0 * NF4 + i) * 16u, qg + i);
      async_copy_b128(base + (uint32_t)(1 * NF4 + i) * 16u, kg + i);
      async_copy_b128(base + (uint32_t)(2 * NF4 + i) * 16u, vg + i);
    }
  }

  // Load initial state into LDS (overlaps with the async chunk-0 fetch).
  for (int i = tid; i < HD_ * HD_; i += 256)
    Wsh[i] = state[(size_t)b * HD_ * HD_ + i];

  for (int c = 0; c < NCH_; ++c) {
    const int buf = c & 1;
    float* Qc = cbuf + buf * 3 * CH_ * HD_;
    float* Kc = Qc + CH_ * HD_;
    float* Vc = Kc + CH_ * HD_;

    // Chunk-c tiles landed; all waves done with the other buffer.
    wait_async0();
    __syncthreads();

    // Kick off the async fetch of chunk c+1 into the other buffer.
    if (c + 1 < NCH_) {
      const size_t src = ((size_t)b * S_ + (size_t)(c + 1) * CH_) * HD_;
      const float4* qg = (const float4*)(Q + src);
      const float4* kg = (const float4*)(K + src);
      const float4* vg = (const float4*)(V + src);
      const uint32_t base = lds_addr_of(cbuf + (buf ^ 1) * 3 * CH_ * HD_);
      for (int i = tid; i < NF4; i += 256) {
        async_copy_b128(base + (uint32_t)(0 * NF4 + i) * 16u, qg + i);
        async_copy_b128(base + (uint32_t)(1 * NF4 + i) * 16u, kg + i);
        async_copy_b128(base + (uint32_t)(2 * NF4 + i) * 16u, vg + i);
      }
    }

    // --- Gram matrices (waves 0 and 1) ---
    if (wv == 0) {
      v8f acc = {};
      for (int k0 = 0; k0 < HD_; k0 += 4) {
        v2f a;  // A = Kc, B = Kc^T  -> identical per-lane fragments
        a.x = Kc[lr16 * HD_ + k0 + 2 * half + 0];
        a.y = Kc[lr16 * HD_ + k0 + 2 * half + 1];
        acc = wmma4(a, a, acc);
      }
#pragma unroll
      for (int r = 0; r < 8; ++r) Ag[(r + 8 * half) * CH_ + lr16] = acc[r];
    } else if (wv == 1) {
      v8f acc = {};
      for (int k0 = 0; k0 < HD_; k0 += 4) {
        v2f a, bq;  // A = Kc, B = Qc^T
        a.x  = Kc[lr16 * HD_ + k0 + 2 * half + 0];
        a.y  = Kc[lr16 * HD_ + k0 + 2 * half + 1];
        bq.x = Qc[lr16 * HD_ + k0 + 2 * half + 0];
        bq.y = Qc[lr16 * HD_ + k0 + 2 * half + 1];
        acc = wmma4(a, bq, acc);
      }
#pragma unroll
      for (int r = 0; r < 8; ++r) Mg[(r + 8 * half) * CH_ + lr16] = acc[r];
    }

    // --- Yq0 / Yk0 for this wave's 16-row slice of W ---
    {
      v8f aq = {}, ak = {};
      for (int k0 = 0; k0 < HD_; k0 += 4) {
        v2f a, bq, bk;
        a.x  = Wsh[(row0 + lr16) * HD_ + k0 + 2 * half + 0];
        a.y  = Wsh[(row0 + lr16) * HD_ + k0 + 2 * half + 1];
        bq.x = Qc[lr16 * HD_ + k0 + 2 * half + 0];
        bq.y = Qc[lr16 * HD_ + k0 + 2 * half + 1];
        bk.x = Kc[lr16 * HD_ + k0 + 2 * half + 0];
        bk.y = Kc[lr16 * HD_ + k0 + 2 * half + 1];
        aq = wmma4(a, bq, aq);
        ak = wmma4(a, bk, ak);
      }
#pragma unroll
      for (int r = 0; r < 8; ++r) {
        yqb[wv * 256 + (r + 8 * half) * CH_ + lr16] = aq[r];
        ykb[wv * 256 + (r + 8 * half) * CH_ + lr16] = ak[r];
      }
    }
    __syncthreads();  // Ag/Mg ready for everyone

    // --- forward substitution: 16 serial steps, wave-local (lanes 0..15) ---
    if (lane < 16) {
      const int r = row0 + lane;  // global W row owned by this lane
      const float* myyq = yqb + wv * 256 + lane * CH_;
      const float* myyk = ykb + wv * 256 + lane * CH_;
      float err[CH_];
#pragma unroll
      for (int t = 0; t < CH_; ++t) {
        float e  = myyk[t] - Vc[t * HD_ + r];  // (W0 k_t - v_t)[r]
        float yy = myyq[t];                    // (W0 q_t)[r]
        for (int s = 0; s < t; ++s) {
          e  -= eta * Ag[s * CH_ + t] * err[s];
          yy -= eta * Mg[s * CH_ + t] * err[s];
        }
        err[t] = e;
        Ys[((size_t)b * S_ + (size_t)c * CH_ + t) * HD_ + r] = yy;
        erb[wv * 256 + lane * CH_ + t] = -eta * e;  // pre-scaled for update
      }
    }
    // LDS ops from the same wave complete in order -> erb visible below.

    // --- rank-16 state update: W_slice += (-eta*Err)(16x16) * Kc(16x128) ---
#pragma unroll
    for (int nt = 0; nt < 8; ++nt) {
      v8f acc;
#pragma unroll
      for (int r = 0; r < 8; ++r)
        acc[r] = Wsh[(row0 + r + 8 * half) * HD_ + nt * 16 + lr16];
#pragma unroll
      for (int k0 = 0; k0 < CH_; k0 += 4) {
        v2f a, bb;
        a.x  = erb[wv * 256 + lr16 * CH_ + k0 + 2 * half + 0];
        a.y  = erb[wv * 256 + lr16 * CH_ + k0 + 2 * half + 1];
        bb.x = Kc[(k0 + 2 * half + 0) * HD_ + nt * 16 + lr16];
        bb.y = Kc[(k0 + 2 * half + 1) * HD_ + nt * 16 + lr16];
        acc = wmma4(a, bb, acc);
      }
#pragma unroll
      for (int r = 0; r < 8; ++r)
        Wsh[(row0 + r + 8 * half) * HD_ + nt * 16 + lr16] = acc[r];
    }
    // No trailing barrier: next iteration's wait_async0 + __syncthreads
    // orders buffer reuse, Ag/Mg rewrite, and W-slice consumption.
  }

  __syncthreads();
  for (int i = tid; i < HD_ * HD_; i += 256)
    Wfinal[(size_t)b * HD_ * HD_ + i] = Wsh[i];
}

// ---------------------------------------------------------------------------
// Kernel 3: out = Ys @ Wo.  grid = M/128, block = 256 (8 waves).
// Each wave: one 16-row M-tile, 8 passes of 8 N-tiles each (N = 1024).
// ---------------------------------------------------------------------------
__global__ void titans_outproj(const float* __restrict__ Ys,
                               const float* __restrict__ Wo,
                               float* __restrict__ out) {
  const int lane = threadIdx.x & 31;
  const int wv   = threadIdx.x >> 5;
  const int half = lane >> 4;
  const int lr   = lane & 15;
  const size_t row0 = (size_t)blockIdx.x * 128 + (size_t)wv * 16;
  const size_t rA   = row0 + (size_t)lr;

  for (int ng = 0; ng < 8; ++ng) {
    v8f acc[8];
#pragma unroll
    for (int i = 0; i < 8; ++i) { v8f z = {}; acc[i] = z; }
    for (int k0 = 0; k0 < HD_; k0 += 4) {
      v2f a;
      a.x = Ys[rA * HD_ + (size_t)(k0 + 2 * half + 0)];
      a.y = Ys[rA * HD_ + (size_t)(k0 + 2 * half + 1)];
#pragma unroll
      for (int nt = 0; nt < 8; ++nt) {
        const int col = (ng * 8 + nt) * 16 + lr;
        v2f b;
        b.x = Wo[(size_t)(k0 + 2 * half + 0) * D_ + col];
        b.y = Wo[(size_t)(k0 + 2 * half + 1) * D_ + col];
        acc[nt] = wmma4(a, b, acc[nt]);
      }
    }
#pragma unroll
    for (int nt = 0; nt < 8; ++nt) {
      const int col = (ng * 8 + nt) * 16 + lr;
#pragma unroll
      for (int r = 0; r < 8; ++r)
        out[(row0 + (size_t)(r + 8 * half)) * D_ + col] = acc[nt][r];
    }
  }
}

// ---------------------------------------------------------------------------
extern "C" void kernel_launch(void* const* d_in, const int* in_sizes, int n_in,
                              void* d_out, int out_size, void* d_ws, size_t ws_size,
                              hipStream_t stream) {
  const float* x  = (const float*)d_in[0];
  const float* Wq = (const float*)d_in[1];
  const float* Wk = (const float*)d_in[2];
  const float* Wv = (const float*)d_in[3];
  const float* Wo = (const float*)d_in[4];
  const float* lr = (const float*)d_in[5];
  const float* st = (const float*)d_in[6];

  float* out  = (float*)d_out;                               // [B,S,D]
  float* Wfin = out + (size_t)B_ * S_ * D_;                  // [B,HD,HD]

  const size_t nqkv = (size_t)B_ * S_ * HD_;                 // 2,097,152 floats
  if (ws_size < 4 * nqkv * sizeof(float)) return;            // need 32 MB scratch
  float* Q  = (float*)d_ws;
  float* Kk = Q + nqkv;
  float* V  = Kk + nqkv;
  float* Ys = V + nqkv;

  // q,k,v projections
  titans_proj<<<dim3((B_ * S_) / 128, 3), 256, 0, stream>>>(x, Wq, Wk, Wv, Q, Kk, V);

  // chunked delta-rule scan (one workgroup per batch, W in LDS, async
  // double-buffered chunk tiles)
  const size_t smem =
      (size_t)(HD_ * HD_ + 2 * 3 * CH_ * HD_ + 2 * CH_ * CH_ + 3 * 8 * CH_ * CH_) *
      sizeof(float);
  titans_scan<<<B_, 256, smem, stream>>>(Q, Kk, V, st, lr, Ys, Wfin);

  // output projection
  titans_outproj<<<(B_ * S_) / 128, 256, 0, stream>>>(Ys, Wo, out);
}